// GVPDynamicProjection_14800457302135
// MI455X (gfx1250) — compile-verified
//
#include <hip/hip_runtime.h>
#include <math.h>

#define N_PTS 10000
#define C_DIM 2
#define SIZ   128
#define VIZ   16
#define H_DIM 4
#define R_DIM 32
#define SH    32
#define VH    4
#define XDIM  144
#define EPSF  1e-8f
#define CHUNKS  50
#define CHUNK_N 200

typedef float v2f __attribute__((ext_vector_type(2)));
typedef float v8f __attribute__((ext_vector_type(8)));

__device__ __forceinline__ float sigmoidf_(float x) { return 1.0f / (1.0f + expf(-x)); }

// ---------------------------------------------------------------------------
// K0: repack a (KD x ND) row-major weight so each WMMA lane's (K,K+1) pair is
// a contiguous, 8B-aligned float2:  P[((g*ND+col)*2+p)] = {W[4g+2p][col], W[4g+2p+1][col]}
// ---------------------------------------------------------------------------
__global__ void k0_pack(const float* __restrict__ W, int KD, int ND, float* __restrict__ P)
{
  int idx = blockIdx.x * blockDim.x + threadIdx.x;  // one float2 entry each
  int total = (KD >> 2) * ND * 2;
  if (idx >= total) return;
  int p = idx & 1;
  int col = (idx >> 1) % ND;
  int g = (idx >> 1) / ND;
  int k = g * 4 + p * 2;
  P[idx * 2 + 0] = W[k * ND + col];
  P[idx * 2 + 1] = W[(k + 1) * ND + col];
}

// ---------------------------------------------------------------------------
// K1: per-point projection logits  pw[n,c,a] = [s, ||v^T wh||] @ wp_ws_w + b
//     144x32 GEMM via WMMA f32 16x16x4 (2 waves, 2 col-tiles)
// ---------------------------------------------------------------------------
__global__ void __launch_bounds__(256) k1_pw_logits(
    const float* __restrict__ s, const float* __restrict__ v,
    const float* __restrict__ wh, const float* __restrict__ wppack,
    const float* __restrict__ wsb, float* __restrict__ logits)
{
  __shared__ float xs[16][XDIM];
  __shared__ float vhl[16][48];
  const int c = blockIdx.y;
  const int n0 = blockIdx.x * 16;
  const int tid = threadIdx.x;

  for (int idx = tid; idx < 16 * SIZ; idx += 256) {
    int row = idx >> 7, col = idx & 127;
    int n = n0 + row;
    xs[row][col] = (n < N_PTS) ? s[(n * C_DIM + c) * SIZ + col] : 0.0f;
  }
  __syncthreads();
  for (int idx = tid; idx < 16 * 48; idx += 256) {
    int row = idx / 48, rem = idx % 48;
    int j = rem / 16, d = rem % 16;
    int n = n0 + row;
    float acc = 0.0f;
    if (n < N_PTS) {
      const float* vp = v + (n * C_DIM + c) * VIZ * 3;
      for (int k = 0; k < VIZ; ++k) acc += vp[k * 3 + j] * wh[k * VIZ + d];
    }
    vhl[row][j * 16 + d] = acc;
  }
  __syncthreads();
  for (int idx = tid; idx < 16 * VIZ; idx += 256) {
    int row = idx >> 4, d = idx & 15;
    float a0 = vhl[row][d], a1 = vhl[row][16 + d], a2 = vhl[row][32 + d];
    xs[row][SIZ + d] = sqrtf(fmaxf(a0 * a0 + a1 * a1 + a2 * a2, EPSF));
  }
  __syncthreads();

  // WMMA: logits(16x32) = x(16x144) @ wp_ws_w(144x32); waves 0-1 fully active
  if (tid < 64) {
    const int lane = tid & 31;
    const int wave = tid >> 5;
    const int colBase = wave * 16;
    const int row = lane & 15;
    const int khalf = (lane >> 4) << 1;
    const v2f* bp = (const v2f*)wppack;
    v8f acc = {};
    for (int g = 0; g < XDIM / 4; ++g) {
      v2f a, b;
      a[0] = xs[row][g * 4 + khalf];
      a[1] = xs[row][g * 4 + khalf + 1];
      b = bp[(g * R_DIM + colBase + row) * 2 + (khalf >> 1)];
      acc = __builtin_amdgcn_wmma_f32_16x16x4_f32(
          false, a, false, b, (short)0, acc, false, false);
    }
    const int hi = (lane >> 4) * 8;
    for (int i = 0; i < 8; ++i) {
      int rown = i + hi;
      int col = colBase + row;
      int n = n0 + rown;
      if (n < N_PTS) logits[(n * C_DIM + c) * R_DIM + col] = acc[i] + wsb[col];
    }
  }
}

// ---------------------------------------------------------------------------
// K2: softmax stats over N for each (c,a)
// ---------------------------------------------------------------------------
__global__ void __launch_bounds__(256) k2_softmax_stats(
    const float* __restrict__ logits, float* __restrict__ smax, float* __restrict__ ssum)
{
  const int ca = blockIdx.x;  // c*R + a
  const int tid = threadIdx.x;
  __shared__ float red[256];
  float m = -1e30f;
  for (int n = tid; n < N_PTS; n += 256) m = fmaxf(m, logits[n * (C_DIM * R_DIM) + ca]);
  red[tid] = m; __syncthreads();
  for (int s2 = 128; s2 > 0; s2 >>= 1) { if (tid < s2) red[tid] = fmaxf(red[tid], red[tid + s2]); __syncthreads(); }
  float mx = red[0]; __syncthreads();
  float acc = 0.0f;
  for (int n = tid; n < N_PTS; n += 256) acc += expf(logits[n * (C_DIM * R_DIM) + ca] - mx);
  red[tid] = acc; __syncthreads();
  for (int s2 = 128; s2 > 0; s2 >>= 1) { if (tid < s2) red[tid] += red[tid + s2]; __syncthreads(); }
  if (tid == 0) { smax[ca] = mx; ssum[ca] = red[0]; }
}

// ---------------------------------------------------------------------------
// K3a: chunked partial weighted sums; K3b: reduce chunks -> s_g, v_g
// ---------------------------------------------------------------------------
__global__ void __launch_bounds__(192) k3a_partial(
    const float* __restrict__ logits, const float* __restrict__ smax,
    const float* __restrict__ s, const float* __restrict__ v,
    float* __restrict__ partials)
{
  const int ca = blockIdx.x;
  const int chunk = blockIdx.y;
  const int c = ca >> 5;
  const int f = threadIdx.x;
  if (f >= SIZ + 48) return;
  const float mx = smax[ca];
  const int nb = chunk * CHUNK_N;
  const int ne = (nb + CHUNK_N < N_PTS) ? nb + CHUNK_N : N_PTS;
  float acc = 0.0f;
  for (int n = nb; n < ne; ++n) {
    float w = expf(logits[n * (C_DIM * R_DIM) + ca] - mx);
    float val = (f < SIZ) ? s[(n * C_DIM + c) * SIZ + f]
                          : v[(n * C_DIM + c) * 48 + (f - SIZ)];
    acc += w * val;
  }
  partials[(ca * CHUNKS + chunk) * 176 + f] = acc;
}

__global__ void __launch_bounds__(192) k3b_reduce(
    const float* __restrict__ partials, const float* __restrict__ ssum,
    float* __restrict__ s_g, float* __restrict__ v_g)
{
  const int ca = blockIdx.x;
  const int c = ca >> 5, a = ca & 31;
  const int f = threadIdx.x;
  if (f >= SIZ + 48) return;
  float acc = 0.0f;
  for (int ch = 0; ch < CHUNKS; ++ch) acc += partials[(ca * CHUNKS + ch) * 176 + f];
  acc /= ssum[ca];
  if (f < SIZ) s_g[(a * C_DIM + c) * SIZ + f] = acc;
  else         v_g[(a * C_DIM + c) * 48 + (f - SIZ)] = acc;
}

// ---------------------------------------------------------------------------
// K4: GVP on the 64 global tokens (k and vv branches)
// ---------------------------------------------------------------------------
__device__ void gvp_token(const float* sg, const float* vg,
                          const float* wh, const float* wsw, const float* wsb,
                          const float* wv, const float* wsvw, const float* wsvb,
                          float* outs, float* outv)
{
  float vh[48];  // [j*16+d]
  for (int j = 0; j < 3; ++j)
    for (int d = 0; d < VIZ; ++d) {
      float acc = 0.0f;
      for (int k = 0; k < VIZ; ++k) acc += vg[k * 3 + j] * wh[k * VIZ + d];
      vh[j * 16 + d] = acc;
    }
  float x[XDIM];
  for (int i = 0; i < SIZ; ++i) x[i] = sg[i];
  for (int d = 0; d < VIZ; ++d) {
    float a0 = vh[d], a1 = vh[16 + d], a2 = vh[32 + d];
    x[SIZ + d] = sqrtf(fmaxf(a0 * a0 + a1 * a1 + a2 * a2, EPSF));
  }
  float so[SIZ];
  for (int o = 0; o < SIZ; ++o) {
    float acc = wsb[o];
    for (int i = 0; i < XDIM; ++i) acc += x[i] * wsw[i * SIZ + o];
    so[o] = acc;
  }
  for (int d = 0; d < VIZ; ++d) {
    float g = wsvb[d];
    for (int o = 0; o < SIZ; ++o) g += sigmoidf_(so[o]) * wsvw[o * VIZ + d];
    g = sigmoidf_(g);
    for (int j = 0; j < 3; ++j) {
      float acc = 0.0f;
      for (int k = 0; k < VIZ; ++k) acc += vh[j * 16 + k] * wv[k * VIZ + d];
      outv[d * 3 + j] = acc * g;
    }
  }
  for (int o = 0; o < SIZ; ++o) outs[o] = so[o] * sigmoidf_(so[o]);
}

__global__ void k4_tokens(
    const float* __restrict__ s_g, const float* __restrict__ v_g,
    const float* __restrict__ k_wh, const float* __restrict__ k_wsw, const float* __restrict__ k_wsb,
    const float* __restrict__ k_wv, const float* __restrict__ k_wsvw, const float* __restrict__ k_wsvb,
    const float* __restrict__ vv_wh, const float* __restrict__ vv_wsw, const float* __restrict__ vv_wsb,
    const float* __restrict__ vv_wv, const float* __restrict__ vv_wsvw, const float* __restrict__ vv_wsvb,
    const float* __restrict__ attn_wh, const float* __restrict__ attn_wsw,
    float* __restrict__ kh_out, float* __restrict__ ek_out,
    float* __restrict__ vs_out, float* __restrict__ vv_out)
{
  int t = threadIdx.x;
  if (t >= R_DIM * C_DIM) return;
  int r = t >> 1, c = t & 1;
  const float* sg = s_g + (r * C_DIM + c) * SIZ;
  const float* vg = v_g + (r * C_DIM + c) * 48;
  float outs[SIZ], outv[48];

  // K branch -> ek (per head), kh (3x8 per head)
  gvp_token(sg, vg, k_wh, k_wsw, k_wsb, k_wv, k_wsvw, k_wsvb, outs, outv);
  const float* wsk = attn_wsw + SH;
  for (int h = 0; h < H_DIM; ++h) {
    float e = 0.0f;
    for (int d = 0; d < SH; ++d) e += outs[h * SH + d] * wsk[d];
    ek_out[(r * C_DIM + c) * H_DIM + h] = e;
    for (int j = 0; j < 3; ++j)
      for (int m = 0; m < 8; ++m) {
        float acc = 0.0f;
        for (int dd = 0; dd < VH; ++dd)
          acc += outv[(h * VH + dd) * 3 + j] * attn_wh[(VH + dd) * 8 + m];
        kh_out[((r * C_DIM + c) * H_DIM + h) * 24 + j * 8 + m] = acc;
      }
  }
  // VV branch -> vs, vv_
  gvp_token(sg, vg, vv_wh, vv_wsw, vv_wsb, vv_wv, vv_wsvw, vv_wsvb, outs, outv);
  for (int o = 0; o < SIZ; ++o) vs_out[(r * C_DIM + c) * SIZ + o] = outs[o];
  for (int f = 0; f < 48; ++f)  vv_out[(r * C_DIM + c) * 48 + f]  = outv[f];
}

// ---------------------------------------------------------------------------
// K5: per-point q-GVP (WMMA f32 16x16x4, packed-B) + attention over R=32
// ---------------------------------------------------------------------------
__global__ void __launch_bounds__(256) k5_main(
    const float* __restrict__ s, const float* __restrict__ v,
    const float* __restrict__ q_wh, const float* __restrict__ qpack, const float* __restrict__ q_wsb,
    const float* __restrict__ q_wv, const float* __restrict__ q_wsvw, const float* __restrict__ q_wsvb,
    const float* __restrict__ attn_wh, const float* __restrict__ attn_wsw, const float* __restrict__ attn_wsb,
    const float* __restrict__ khg, const float* __restrict__ ekg,
    const float* __restrict__ vsg, const float* __restrict__ vvg,
    float* __restrict__ out_s, float* __restrict__ out_v)
{
  __shared__ float xs[16][XDIM];
  __shared__ float vhl[16][48];
  __shared__ float sol[16][SIZ];
  __shared__ float gatel[16][VIZ];
  __shared__ float qvl[16][48];
  __shared__ float khl[R_DIM][96];
  __shared__ float ekl[R_DIM][H_DIM];
  __shared__ float vsl[R_DIM][SIZ];
  __shared__ float vvl[R_DIM][48];

  const int c = blockIdx.y;
  const int n0 = blockIdx.x * 16;
  const int tid = threadIdx.x;

  for (int idx = tid; idx < R_DIM * 96; idx += 256) {
    int r = idx / 96, rem = idx % 96;
    khl[r][rem] = khg[(r * C_DIM + c) * 96 + rem];
  }
  for (int idx = tid; idx < R_DIM * H_DIM; idx += 256) {
    int r = idx / H_DIM, h = idx % H_DIM;
    ekl[r][h] = ekg[(r * C_DIM + c) * H_DIM + h];
  }
  for (int idx = tid; idx < R_DIM * SIZ; idx += 256) {
    int r = idx >> 7, o = idx & 127;
    vsl[r][o] = vsg[(r * C_DIM + c) * SIZ + o];
  }
  for (int idx = tid; idx < R_DIM * 48; idx += 256) {
    int r = idx / 48, f = idx % 48;
    vvl[r][f] = vvg[(r * C_DIM + c) * 48 + f];
  }
  for (int idx = tid; idx < 16 * SIZ; idx += 256) {
    int row = idx >> 7, col = idx & 127;
    int n = n0 + row;
    xs[row][col] = (n < N_PTS) ? s[(n * C_DIM + c) * SIZ + col] : 0.0f;
  }
  __syncthreads();
  for (int idx = tid; idx < 16 * 48; idx += 256) {
    int row = idx / 48, rem = idx % 48;
    int j = rem / 16, d = rem % 16;
    int n = n0 + row;
    float acc = 0.0f;
    if (n < N_PTS) {
      const float* vp = v + (n * C_DIM + c) * 48;
      for (int k = 0; k < VIZ; ++k) acc += vp[k * 3 + j] * q_wh[k * VIZ + d];
    }
    vhl[row][j * 16 + d] = acc;
  }
  __syncthreads();
  for (int idx = tid; idx < 16 * VIZ; idx += 256) {
    int row = idx >> 4, d = idx & 15;
    float a0 = vhl[row][d], a1 = vhl[row][16 + d], a2 = vhl[row][32 + d];
    xs[row][SIZ + d] = sqrtf(fmaxf(a0 * a0 + a1 * a1 + a2 * a2, EPSF));
  }
  __syncthreads();

  // === WMMA f32 16x16x4: so(16x128) = x(16x144) @ q_wsw(144x128) ===
  {
    const int lane = tid & 31;
    const int wave = tid >> 5;
    const int colBase = wave * 16;           // 8 waves x 16 cols = 128
    const int row = lane & 15;               // A: M = row
    const int khalf = (lane >> 4) << 1;      // lanes 16-31 carry K+2,K+3
    const v2f* bp = (const v2f*)qpack;
    v8f acc = {};
    for (int g = 0; g < XDIM / 4; ++g) {
      v2f a, b;
      a[0] = xs[row][g * 4 + khalf];
      a[1] = xs[row][g * 4 + khalf + 1];
      b = bp[(g * SIZ + colBase + row) * 2 + (khalf >> 1)];
      acc = __builtin_amdgcn_wmma_f32_16x16x4_f32(
          false, a, false, b, (short)0, acc, false, false);
    }
    const int hi = (lane >> 4) * 8;          // C/D layout: VGPR i -> M = i (+8)
    for (int i = 0; i < 8; ++i) {
      int rown = i + hi;
      int col = colBase + row;
      sol[rown][col] = acc[i] + q_wsb[col];
    }
  }
  __syncthreads();

  // gate = sigmoid(sigmoid(so) @ wsv_w + b)
  for (int idx = tid; idx < 16 * VIZ; idx += 256) {
    int row = idx >> 4, d = idx & 15;
    float g = q_wsvb[d];
    for (int o = 0; o < SIZ; ++o) g += sigmoidf_(sol[row][o]) * q_wsvw[o * VIZ + d];
    gatel[row][d] = sigmoidf_(g);
  }
  __syncthreads();
  // qv = (vh @ q_wv)^T * gate
  for (int idx = tid; idx < 16 * 48; idx += 256) {
    int row = idx / 48, rem = idx % 48;
    int d = rem / 3, j = rem % 3;
    float acc = 0.0f;
    for (int k = 0; k < VIZ; ++k) acc += vhl[row][j * 16 + k] * q_wv[k * VIZ + d];
    qvl[row][rem] = acc * gatel[row][d];
  }
  __syncthreads();

  // attention over R=32 global tokens: one thread per (point-row, head)
  if (tid < 16 * H_DIM) {
    int row = tid >> 2, h = tid & 3;
    int n = n0 + row;
    if (n < N_PTS) {
      float qh[24];  // [j*8+m]
      for (int j = 0; j < 3; ++j)
        for (int m = 0; m < 8; ++m) {
          float acc = 0.0f;
          for (int dd = 0; dd < VH; ++dd)
            acc += qvl[row][(h * VH + dd) * 3 + j] * attn_wh[dd * 8 + m];
          qh[j * 8 + m] = acc;
        }
      float eq = attn_wsb[0];
      for (int d = 0; d < SH; ++d) {
        float so = sol[row][h * SH + d];
        eq += so * sigmoidf_(so) * attn_wsw[d];  // ws_q
      }
      const float* wsn = attn_wsw + 2 * SH;
      const float scale = 0.17677669529663689f;  // 1/sqrt(32)
      float e[R_DIM];
      float mx = -1e30f;
      for (int r = 0; r < R_DIM; ++r) {
        float ev = eq + ekl[r][h];
        for (int m = 0; m < 8; ++m) {
          float s0 = qh[m]      + khl[r][h * 24 + m];
          float s1 = qh[8 + m]  + khl[r][h * 24 + 8 + m];
          float s2 = qh[16 + m] + khl[r][h * 24 + 16 + m];
          ev += wsn[m] * sqrtf(fmaxf(s0 * s0 + s1 * s1 + s2 * s2, EPSF));
        }
        ev *= scale;
        e[r] = ev;
        mx = fmaxf(mx, ev);
      }
      float sum = 0.0f;
      for (int r = 0; r < R_DIM; ++r) { e[r] = expf(e[r] - mx); sum += e[r]; }
      float inv = 1.0f / sum;
      float* osp = out_s + (n * C_DIM + c) * SIZ + h * SH;
      for (int d = 0; d < SH; ++d) {
        float acc = 0.0f;
        for (int r = 0; r < R_DIM; ++r) acc += e[r] * vsl[r][h * SH + d];
        osp[d] = acc * inv;
      }
      float* ovp = out_v + (n * C_DIM + c) * 48 + h * VH * 3;
      for (int f = 0; f < VH * 3; ++f) {
        float acc = 0.0f;
        for (int r = 0; r < R_DIM; ++r) acc += e[r] * vvl[r][h * VH * 3 + f];
        ovp[f] = acc * inv;
      }
    }
  }
}

// ---------------------------------------------------------------------------
extern "C" void kernel_launch(void* const* d_in, const int* in_sizes, int n_in,
                              void* d_out, int out_size, void* d_ws, size_t ws_size,
                              hipStream_t stream) {
  (void)in_sizes; (void)n_in; (void)out_size; (void)ws_size;
  const float* s       = (const float*)d_in[0];
  const float* v       = (const float*)d_in[1];
  const float* wp_wh   = (const float*)d_in[2];
  const float* wp_wsw  = (const float*)d_in[3];
  const float* wp_wsb  = (const float*)d_in[4];
  const float* q_wh    = (const float*)d_in[5];
  const float* q_wsw   = (const float*)d_in[6];
  const float* q_wsb   = (const float*)d_in[7];
  const float* q_wv    = (const float*)d_in[8];
  const float* q_wsvw  = (const float*)d_in[9];
  const float* q_wsvb  = (const float*)d_in[10];
  const float* k_wh    = (const float*)d_in[11];
  const float* k_wsw   = (const float*)d_in[12];
  const float* k_wsb   = (const float*)d_in[13];
  const float* k_wv    = (const float*)d_in[14];
  const float* k_wsvw  = (const float*)d_in[15];
  const float* k_wsvb  = (const float*)d_in[16];
  const float* vv_wh   = (const float*)d_in[17];
  const float* vv_wsw  = (const float*)d_in[18];
  const float* vv_wsb  = (const float*)d_in[19];
  const float* vv_wv   = (const float*)d_in[20];
  const float* vv_wsvw = (const float*)d_in[21];
  const float* vv_wsvb = (const float*)d_in[22];
  const float* attn_wh  = (const float*)d_in[23];
  const float* attn_wsw = (const float*)d_in[24];
  const float* attn_wsb = (const float*)d_in[25];

  float* ws = (float*)d_ws;
  float* logits   = ws;                                        // 640000
  float* smax     = logits + (size_t)N_PTS * C_DIM * R_DIM;    // 64
  float* ssum     = smax + C_DIM * R_DIM;                      // 64
  float* s_g      = ssum + C_DIM * R_DIM;                      // 8192
  float* v_g      = s_g + R_DIM * C_DIM * SIZ;                 // 3072
  float* kh       = v_g + R_DIM * C_DIM * 48;                  // 6144
  float* ek       = kh + R_DIM * C_DIM * H_DIM * 24;           // 256
  float* vs       = ek + R_DIM * C_DIM * H_DIM;                // 8192
  float* vvb      = vs + R_DIM * C_DIM * SIZ;                  // 3072
  float* qpack    = vvb + R_DIM * C_DIM * 48;                  // 18432
  float* wppack   = qpack + XDIM * SIZ;                        // 4608
  float* partials = wppack + XDIM * R_DIM;                     // 64*50*176

  float* out_s = (float*)d_out;
  float* out_v = out_s + (size_t)N_PTS * C_DIM * SIZ;

  const int nblk = (N_PTS + 15) / 16;  // 625

  {
    int tq = (XDIM >> 2) * SIZ * 2;
    k0_pack<<<(tq + 255) / 256, 256, 0, stream>>>(q_wsw, XDIM, SIZ, qpack);
    int tp = (XDIM >> 2) * R_DIM * 2;
    k0_pack<<<(tp + 255) / 256, 256, 0, stream>>>(wp_wsw, XDIM, R_DIM, wppack);
  }
  k1_pw_logits<<<dim3(nblk, C_DIM), 256, 0, stream>>>(s, v, wp_wh, wppack, wp_wsb, logits);
  k2_softmax_stats<<<C_DIM * R_DIM, 256, 0, stream>>>(logits, smax, ssum);
  k3a_partial<<<dim3(C_DIM * R_DIM, CHUNKS), 192, 0, stream>>>(logits, smax, s, v, partials);
  k3b_reduce<<<C_DIM * R_DIM, 192, 0, stream>>>(partials, ssum, s_g, v_g);
  k4_tokens<<<1, 64, 0, stream>>>(s_g, v_g,
                                  k_wh, k_wsw, k_wsb, k_wv, k_wsvw, k_wsvb,
                                  vv_wh, vv_wsw, vv_wsb, vv_wv, vv_wsvw, vv_wsvb,
                                  attn_wh, attn_wsw, kh, ek, vs, vvb);
  k5_main<<<dim3(nblk, C_DIM), 256, 0, stream>>>(s, v,
                                                 q_wh, qpack, q_wsb, q_wv, q_wsvw, q_wsvb,
                                                 attn_wh, attn_wsw, attn_wsb,
                                                 kh, ek, vs, vvb, out_s, out_v);
}